// GlobalAttentionLayer_10075993276619
// MI455X (gfx1250) — compile-verified
//
#include <hip/hip_runtime.h>
#include <hip/hip_bf16.h>

typedef float v2f __attribute__((ext_vector_type(2)));
typedef float v8f __attribute__((ext_vector_type(8)));

#define D_FEAT 512
#define ROWS_PER_BLOCK 128   // 8 waves * 16 rows
#define KTILE 32
#define AP 36                // padded LDS row stride (words): 16B-aligned rows, conflict-free frags

#if __has_builtin(__builtin_amdgcn_s_wait_asynccnt)
#define WAIT_ASYNC(n) __builtin_amdgcn_s_wait_asynccnt(n)
#else
#define WAIT_ASYNC(n) asm volatile("s_wait_asynccnt %0" ::"i"(n) : "memory")
#endif

// ---- monotone float<->uint encoding so unsigned atomicMax orders like float max ----
__device__ __forceinline__ unsigned f2ord(float f) {
    unsigned u = __float_as_uint(f);
    return (u & 0x80000000u) ? ~u : (u | 0x80000000u);
}
__device__ __forceinline__ float ord2f(unsigned u) {
    return (u & 0x80000000u) ? __uint_as_float(u & 0x7fffffffu) : __uint_as_float(~u);
}

// Issue one 128x32 f32 tile as async global->LDS B128 copies.
// Per thread: 4 instructions (rows rr, rr+32, rr+64, rr+96 at column c4*4).
// ASYNCcnt += 4 per wave per tile.
__device__ __forceinline__ void issue_tile_async(const float* gbase, const float* lds_base,
                                                 int rr, int c4) {
    const float* gp = gbase + (long)rr * D_FEAT + c4 * 4;
    unsigned lo = (unsigned)(size_t)(lds_base + rr * AP + c4 * 4);  // low 32b = LDS byte offset
#pragma unroll
    for (int s = 0; s < 4; ++s) {
        asm volatile("global_load_async_to_lds_b128 %0, %1, off" ::"v"(lo), "v"(gp)
                     : "memory");
        gp += 32 * D_FEAT;
        lo += 32 * AP * 4;
    }
}

// ---------------------------------------------------------------------------
// Kernel 0: exclusive prefix sum of graph_sizes -> offs[0..B]; init max slot.
// ---------------------------------------------------------------------------
__global__ void scan_offsets(const int* __restrict__ sizes, int B,
                             int* __restrict__ offs, unsigned* __restrict__ maxslot) {
    __shared__ int s[1024];
    int tid = threadIdx.x;
    s[tid] = (tid < B) ? sizes[tid] : 0;
    __syncthreads();
#pragma unroll
    for (int d = 1; d < 1024; d <<= 1) {
        int t = (tid >= d) ? s[tid - d] : 0;
        __syncthreads();
        s[tid] += t;
        __syncthreads();
    }
    if (tid == 0) {
        offs[0] = 0;
        maxslot[0] = 0x007fffffu;  // f2ord(-inf)
    }
    if (tid < B) offs[tid + 1] = s[tid];
}

// ---------------------------------------------------------------------------
// Kernel 1: streaming GEMV via V_WMMA_F32_16X16X4_F32 with double-buffered
// async global->LDS staging (ASYNCcnt pipelined against the WMMA loop).
// ---------------------------------------------------------------------------
__global__ void gemv_wmma(const float* __restrict__ inp,
                          const float* __restrict__ Wg, const float* __restrict__ bg,
                          const float* __restrict__ Wo, const float* __restrict__ bo,
                          float* __restrict__ g, float* __restrict__ o0,
                          float* __restrict__ o1) {
    __shared__ __attribute__((aligned(16))) float Atile[2][ROWS_PER_BLOCK * AP];  // 2x18432 B
    __shared__ float Bc[D_FEAT * 3];                                              // 6144 B

    const int tid  = threadIdx.x;
    const int wave = tid >> 5;
    const int lane = tid & 31;
    const int n    = lane & 15;   // N index within 16-wide tile
    const bool hi  = lane >= 16;  // hi half supplies K+2/K+3
    const int koff = hi ? 2 : 0;
    const int arow = (wave << 4) + n;
    const long node0 = (long)blockIdx.x * ROWS_PER_BLOCK;
    const float* gbase = inp + node0 * D_FEAT;
    const int rr = tid >> 3;  // 0..31: row within tile for the async copy
    const int c4 = tid & 7;   // 16-B column slot

    // Prologue: kick off tile 0 while we build the compact B table.
    issue_tile_async(gbase, &Atile[0][0], rr, c4);

    for (int i = tid; i < D_FEAT; i += 256) {
        Bc[i * 3 + 0] = Wg[i];
        Bc[i * 3 + 1] = Wo[2 * i + 0];
        Bc[i * 3 + 2] = Wo[2 * i + 1];
    }

    v8f acc = {};

    for (int kc = 0; kc < D_FEAT; kc += KTILE) {
        const int cur = (kc >> 5) & 1;
        if (kc + KTILE < D_FEAT) {
            issue_tile_async(gbase + (kc + KTILE), &Atile[cur ^ 1][0], rr, c4);
            WAIT_ASYNC(4);  // current tile landed; next tile still in flight
        } else {
            WAIT_ASYNC(0);
        }
        __syncthreads();  // current tile visible to all waves (Bc too, on iter 0)

        const float* At = &Atile[cur][0];
#pragma unroll
        for (int kk = 0; kk < KTILE; kk += 4) {
            const float* ap = &At[arow * AP + kk + koff];
            v2f a;
            a.x = ap[0];
            a.y = ap[1];
            v2f bfrag;
            int kb = kc + kk + koff;
            if (n < 3) {
                bfrag.x = Bc[kb * 3 + n];
                bfrag.y = Bc[(kb + 1) * 3 + n];
            } else {
                bfrag.x = 0.0f;
                bfrag.y = 0.0f;
            }
            acc = __builtin_amdgcn_wmma_f32_16x16x4_f32(
                false, a, false, bfrag, (short)0, acc, false, false);
        }
        __syncthreads();  // all waves done with `cur` before it is re-filled (WAR)
    }

    // Scatter: lanes with column n in {0,1,2} hold gate / out0 / out1.
    const float bg0 = bg[0], bo0 = bo[0], bo1 = bo[1];
    if (n < 3) {
#pragma unroll
        for (int r = 0; r < 8; ++r) {
            int m = (hi ? 8 : 0) + r;  // C/D layout: VGPR r -> M=r / M=8+r
            long node = node0 + (long)(wave << 4) + m;
            float v = acc[r];
            if (n == 0)
                g[node] = v + bg0;
            else if (n == 1)
                o0[node] = v + bo0;
            else
                o1[node] = v + bo1;
        }
    }
}

// ---------------------------------------------------------------------------
// Kernel 2: global max of g (reference subtracts the GLOBAL max).
// ---------------------------------------------------------------------------
__global__ void reduce_max(const float* __restrict__ g, long nelem,
                           unsigned* __restrict__ maxslot) {
    __shared__ float red[256];
    long i      = (long)blockIdx.x * 256 + threadIdx.x;
    long stride = (long)gridDim.x * 256;
    float m = -__builtin_inff();
    for (; i < nelem; i += stride) m = fmaxf(m, g[i]);
    red[threadIdx.x] = m;
    __syncthreads();
    for (int d = 128; d > 0; d >>= 1) {
        if (threadIdx.x < d) red[threadIdx.x] = fmaxf(red[threadIdx.x], red[threadIdx.x + d]);
        __syncthreads();
    }
    if (threadIdx.x == 0) atomicMax(maxslot, f2ord(red[0]));
}

// ---------------------------------------------------------------------------
// Kernel 3: per-graph softmax-normalized weighted readout. One block / graph.
// ---------------------------------------------------------------------------
__global__ void finalize(const float* __restrict__ g, const float* __restrict__ o0,
                         const float* __restrict__ o1, const int* __restrict__ offs,
                         const unsigned* __restrict__ maxslot, float* __restrict__ out) {
    __shared__ float red[256];
    const int b = blockIdx.x, tid = threadIdx.x;
    const int s0 = offs[b], s1 = offs[b + 1];
    const float M = ord2f(*maxslot);

    float lsum = 0.0f;
    for (int i = s0 + tid; i < s1; i += 256) lsum += expf(g[i] - M);
    red[tid] = lsum;
    __syncthreads();
    for (int d = 128; d > 0; d >>= 1) {
        if (tid < d) red[tid] += red[tid + d];
        __syncthreads();
    }
    const float inv = 1.0f / (red[0] + 1e-16f);
    __syncthreads();

    float a0 = 0.0f, a1 = 0.0f;
    for (int i = s0 + tid; i < s1; i += 256) {
        float w = expf(g[i] - M) * inv;
        a0 += w * o0[i];
        a1 += w * o1[i];
    }
    red[tid] = a0;
    __syncthreads();
    for (int d = 128; d > 0; d >>= 1) {
        if (tid < d) red[tid] += red[tid + d];
        __syncthreads();
    }
    const float r0 = red[0];
    __syncthreads();
    red[tid] = a1;
    __syncthreads();
    for (int d = 128; d > 0; d >>= 1) {
        if (tid < d) red[tid] += red[tid + d];
        __syncthreads();
    }
    if (tid == 0) {
        out[2 * b + 0] = r0;
        out[2 * b + 1] = red[0];
    }
}

extern "C" void kernel_launch(void* const* d_in, const int* in_sizes, int n_in,
                              void* d_out, int out_size, void* d_ws, size_t ws_size,
                              hipStream_t stream) {
    const float* inp = (const float*)d_in[0];
    const int*   gs  = (const int*)d_in[1];
    const float* Wg  = (const float*)d_in[2];
    const float* bg  = (const float*)d_in[3];
    const float* Wo  = (const float*)d_in[4];
    const float* bo  = (const float*)d_in[5];
    float* out = (float*)d_out;

    const long N = (long)in_sizes[0] / D_FEAT;  // 262144
    const int  B = in_sizes[1];                 // 1024

    // Workspace layout: g[N] | o0[N] | o1[N] | offs[B+1] | maxslot
    float*    g       = (float*)d_ws;
    float*    o0      = g + N;
    float*    o1      = o0 + N;
    int*      offs    = (int*)(o1 + N);
    unsigned* maxslot = (unsigned*)(offs + (B + 1));

    scan_offsets<<<1, 1024, 0, stream>>>(gs, B, offs, maxslot);
    gemv_wmma<<<(int)(N / ROWS_PER_BLOCK), 256, 0, stream>>>(inp, Wg, bg, Wo, bo, g, o0, o1);
    reduce_max<<<1024, 256, 0, stream>>>(g, N, maxslot);
    finalize<<<B, 256, 0, stream>>>(g, o0, o1, offs, maxslot, out);
}